// DecoderLayer_85968065397055
// MI455X (gfx1250) — compile-verified
//
#include <hip/hip_runtime.h>
#include <math.h>

// Problem constants (from reference)
#define MMATS    2048
#define KPROB    512
#define FANIN    16
#define INTERACT 8
#define PRESERVE 4
#define TOPK     12
#define THRESH   8
#define NPAIRS   28
#define LDSROW   65              // padded row stride (dwords) -> conflict-free
#define MATSZ    (64 * LDSROW)   // 4160 floats per LDS matrix

typedef __attribute__((ext_vector_type(2))) float v2f;
typedef __attribute__((ext_vector_type(8))) float v8f;
typedef __attribute__((ext_vector_type(4))) int   v4i;

#define GLOBAL_AS __attribute__((address_space(1)))
#define LDS_AS    __attribute__((address_space(3)))

// pair index of (a, a+1) in the triu_indices(8, k=1) row-major enumeration
__device__ __constant__ int c_pbase[7] = {0, 7, 13, 18, 22, 25, 27};

// ---------------------------------------------------------------------------
// Kernel 1: log|det| of every 64x64 matrix via LU with partial pivoting.
// One 64-thread block per matrix; matrix lives in LDS (stride-65 rows).
// ---------------------------------------------------------------------------
__global__ void __launch_bounds__(64)
logabsdet_kernel(const float* __restrict__ prev, float* __restrict__ lad) {
    __shared__ float A[64 * LDSROW];
    __shared__ float colv[64];
    __shared__ float s_log;
    __shared__ int   s_piv;

    const int m = blockIdx.x;
    const int t = threadIdx.x;
    const float* src = prev + (size_t)m * 4096;

    for (int e = t; e < 4096; e += 64)
        A[(e >> 6) * LDSROW + (e & 63)] = src[e];
    if (t == 0) s_log = 0.0f;
    __syncthreads();

    for (int j = 0; j < 64; ++j) {
        colv[t] = (t >= j) ? fabsf(A[t * LDSROW + j]) : -1.0f;
        __syncthreads();
        if (t == 0) {
            int piv = j; float best = colv[j];
            for (int i = j + 1; i < 64; ++i)
                if (colv[i] > best) { best = colv[i]; piv = i; }
            s_piv = piv;
            s_log += logf(best);          // best==0 -> -inf (singular)
        }
        __syncthreads();
        const int piv = s_piv;
        if (piv != j) {                   // swap rows j<->piv; thread t owns col t
            float tmp = A[j * LDSROW + t];
            A[j * LDSROW + t] = A[piv * LDSROW + t];
            A[piv * LDSROW + t] = tmp;
        }
        __syncthreads();
        if (t > j) {                      // eliminate; thread t owns row t
            float f = A[t * LDSROW + j] / A[j * LDSROW + j];
            for (int c = j + 1; c < 64; ++c)
                A[t * LDSROW + c] -= f * A[j * LDSROW + c];
        }
        __syncthreads();
    }
    if (t == 0) lad[m] = s_log;
}

// ---------------------------------------------------------------------------
// Kernel 2: one 256-thread block (8 waves) per k.
//   result = sum_a  A_a @ ( sum_{b>a} v_ab * A_b )  + preserve + bias, swish.
// Wave w owns output tiles (ti = w>>1, tj = (w&1)*2 .. +1); A fragment shared
// across the two N tiles. A matrices staged via async global->LDS copies.
// ---------------------------------------------------------------------------
__global__ void __launch_bounds__(256)
fused_kernel(const float* __restrict__ prev,            // (2048,64,64)
             const unsigned char* __restrict__ act,     // (2048,) bool
             const int* __restrict__ pidx,              // (512,16)
             const float* __restrict__ w1,              // (512,64,32)
             const float* __restrict__ b1,              // (512,64)
             const float* __restrict__ w2,              // (512,64)
             const float* __restrict__ b2,              // (512,)
             const float* __restrict__ lad,             // (2048,)
             float* __restrict__ out)                   // 512*4096 + 512
{
    extern __shared__ float smem[];        // 8 mats + 1 W buffer = 9*MATSZ floats
    __shared__ float s_score[FANIN];
    __shared__ int   s_par[FANIN];
    __shared__ int   s_sel[TOPK];
    __shared__ float s_v[32];
    __shared__ float s_c;
    __shared__ int   s_gate;

    const int k    = blockIdx.x;
    const int tid  = threadIdx.x;
    const int lane = tid & 31;
    const int wv   = tid >> 5;
    float* Wbuf = &smem[INTERACT * MATSZ];

    // Stage 1: parent ids, flags, scores
    if (tid < FANIN) {
        int p = pidx[k * FANIN + tid];
        s_par[tid] = p;
        s_score[tid] = act[p] ? lad[p] : -INFINITY;
    }
    __syncthreads();

    // Stage 2: gate + top-12 selection (top_k tie order: first index wins)
    if (tid == 0) {
        int cnt = 0;
        for (int j = 0; j < FANIN; ++j) cnt += act[s_par[j]] ? 1 : 0;
        s_gate = (cnt >= THRESH) ? 1 : 0;
        float sc[FANIN]; int used[FANIN];
        for (int j = 0; j < FANIN; ++j) { sc[j] = s_score[j]; used[j] = 0; }
        for (int s = 0; s < TOPK; ++s) {
            int best = -1; float bv = 0.0f;
            for (int j = 0; j < FANIN; ++j)
                if (!used[j] && (best < 0 || sc[j] > bv)) { best = j; bv = sc[j]; }
            used[best] = 1;
            s_sel[s] = s_par[best];
        }
    }
    __syncthreads();

    if (!s_gate) {          // block-uniform: gated-off k writes zeros, skips all work
        for (int e = tid; e < 4096; e += 256)
            out[(size_t)k * 4096 + e] = 0.0f;
        if (tid == 0) out[(size_t)KPROB * 4096 + k] = 0.0f;
        return;
    }

    // Stage 3a: fold the two linear layers: v = w2*w1 (32), c = w2*b1 + b2
    if (tid < 32) {
        float acc = 0.0f;
        for (int h = 0; h < 64; ++h)
            acc += w2[k * 64 + h] * w1[(size_t)(k * 64 + h) * 32 + tid];
        s_v[tid] = acc;
    } else if (tid == 32) {
        float acc = 0.0f;
        for (int h = 0; h < 64; ++h) acc += w2[k * 64 + h] * b1[k * 64 + h];
        s_c = acc + b2[k];
    }

    // Stage 3b: stage the 8 interact matrices into padded LDS.
    // Prefer CDNA5 async global->LDS (ASYNCcnt-tracked, bypasses VGPRs).
#if __has_builtin(__builtin_amdgcn_global_load_async_to_lds_b128)
    for (int ch = tid; ch < INTERACT * 1024; ch += 256) {   // 16B chunks
        const int j  = ch >> 10;
        const int e  = ch & 1023;
        const int r  = e >> 4;
        const int c4 = (e & 15) << 2;
        const float* g = prev + (size_t)s_sel[j] * 4096 + r * 64 + c4;
        float* l = &smem[j * MATSZ + r * LDSROW + c4];
        __builtin_amdgcn_global_load_async_to_lds_b128(
            (GLOBAL_AS v4i*)g, (LDS_AS v4i*)l, 0, 0);
    }
#if __has_builtin(__builtin_amdgcn_s_wait_asynccnt)
    __builtin_amdgcn_s_wait_asynccnt(0);
#else
    asm volatile("s_wait_asynccnt 0x0" ::: "memory");
#endif
#else
    for (int e = tid; e < INTERACT * 4096; e += 256) {
        int j = e >> 12, r = (e >> 6) & 63, c = e & 63;
        smem[j * MATSZ + r * LDSROW + c] = prev[(size_t)s_sel[j] * 4096 + r * 64 + c];
    }
#endif
    __syncthreads();

    // Stage 4: for each a, build W_a = sum_{b>a} v_ab * A_b, then WMMA A_a @ W_a
    const int ti     = wv >> 1;          // output M tile (0..3)
    const int tjBase = (wv & 1) * 2;     // output N tile pair
    const int half   = lane >> 4;        // K sub-pair select (0: K0/K1, 1: K2/K3)
    const int l16    = lane & 15;
    const int mrow   = ti * 16 + l16;    // A-fragment row for this lane

    v8f acc0 = {};
    v8f acc1 = {};
    for (int a = 0; a < 7; ++a) {
        // build W (all 256 threads, scale folded here instead of per-WMMA)
        const int pb = c_pbase[a];
        for (int e = tid; e < 4096; e += 256) {
            const int r = e >> 6, c = e & 63;
            float acc = 0.0f;
            for (int b = a + 1; b < 8; ++b)
                acc += s_v[pb + (b - a - 1)] * smem[b * MATSZ + r * LDSROW + c];
            Wbuf[r * LDSROW + c] = acc;
        }
        __syncthreads();

        const float* Ma = &smem[a * MATSZ];
        for (int kb = 0; kb < 16; ++kb) {          // K=64 in steps of 4
            const int k0 = kb * 4 + half * 2;
            v2f av;
            av.x = Ma[mrow * LDSROW + k0];
            av.y = Ma[mrow * LDSROW + k0 + 1];
            v2f b0, b1v;
            b0.x  = Wbuf[k0 * LDSROW + tjBase * 16 + l16];
            b0.y  = Wbuf[(k0 + 1) * LDSROW + tjBase * 16 + l16];
            b1v.x = Wbuf[k0 * LDSROW + (tjBase + 1) * 16 + l16];
            b1v.y = Wbuf[(k0 + 1) * LDSROW + (tjBase + 1) * 16 + l16];
            acc0 = __builtin_amdgcn_wmma_f32_16x16x4_f32(
                       false, av, false, b0, (short)0, acc0, false, false);
            acc1 = __builtin_amdgcn_wmma_f32_16x16x4_f32(
                       false, av, false, b1v, (short)0, acc1, false, false);
        }
        __syncthreads();   // protect W before next rebuild
    }

    // Stage 5: + preserve terms + bias, swish, store (C layout: VGPR r -> M=r(+8))
    const float cbias = s_c;
    for (int t = 0; t < 2; ++t) {
        v8f acc = t ? acc1 : acc0;
        const int coln = (tjBase + t) * 16 + l16;
        for (int r = 0; r < 8; ++r) {
            const int rown = ti * 16 + half * 8 + r;
            float x = acc[r];
            for (int q = 0; q < PRESERVE; ++q)
                x += s_v[NPAIRS + q] *
                     prev[(size_t)s_sel[INTERACT + q] * 4096 + rown * 64 + coln];
            x += cbias;
            out[(size_t)k * 4096 + rown * 64 + coln] = x / (1.0f + expf(-x));
        }
    }
    if (tid == 0) out[(size_t)KPROB * 4096 + k] = 1.0f;
}

// ---------------------------------------------------------------------------
extern "C" void kernel_launch(void* const* d_in, const int* in_sizes, int n_in,
                              void* d_out, int out_size, void* d_ws, size_t ws_size,
                              hipStream_t stream) {
    const float*         prev = (const float*)d_in[0];
    const unsigned char* actv = (const unsigned char*)d_in[1];
    const int*           pidx = (const int*)d_in[2];
    const float*         w1   = (const float*)d_in[3];
    const float*         b1   = (const float*)d_in[4];
    const float*         w2   = (const float*)d_in[5];
    const float*         b2   = (const float*)d_in[6];
    float* out = (float*)d_out;
    float* lad = (float*)d_ws;                    // 2048 floats of scratch

    logabsdet_kernel<<<MMATS, 64, 0, stream>>>(prev, lad);

    // 8 interact matrices + 1 weighted-sum buffer, stride-65 rows: ~150 KB LDS
    const size_t shmem = (size_t)(INTERACT + 1) * MATSZ * sizeof(float);
    fused_kernel<<<KPROB, 256, shmem, stream>>>(prev, actv, pidx, w1, b1, w2, b2,
                                                lad, out);
}